// CausalReadout_complex_44667659878942
// MI455X (gfx1250) — compile-verified
//
#include <hip/hip_runtime.h>
#include <cstdint>

#define NN   50000
#define EE   800000
#define DD   128
#define GG   256
#define OUTD 10
#define BNEPS 1e-5f

typedef __attribute__((ext_vector_type(2))) float v2f;
typedef __attribute__((ext_vector_type(8))) float v8f;

__device__ __forceinline__ void atomAddF(float* p, float v) {
  __hip_atomic_fetch_add(p, v, __ATOMIC_RELAXED, __HIP_MEMORY_SCOPE_AGENT);
}

__global__ void zerof_kernel(float* __restrict__ p, long long n) {
  long long i = (long long)blockIdx.x * blockDim.x + threadIdx.x;
  if (i < n) p[i] = 0.f;
}

// ---- node attention: softmax(h @ Wn + bn), write xc = a0*h, xo = a1*h ----
__global__ void node_att_kernel(const float* __restrict__ h,
                                const float* __restrict__ Wn,
                                const float* __restrict__ bn,
                                float* __restrict__ xc, float* __restrict__ xo) {
  int i = blockIdx.x * 8 + (threadIdx.x >> 5);
  int lane = threadIdx.x & 31;
  if (i >= NN) return;
  float4 a = ((const float4*)(h + (size_t)i * DD))[lane];
  int k0 = lane * 4;
  float p0 = a.x*Wn[(k0+0)*2+0] + a.y*Wn[(k0+1)*2+0] + a.z*Wn[(k0+2)*2+0] + a.w*Wn[(k0+3)*2+0];
  float p1 = a.x*Wn[(k0+0)*2+1] + a.y*Wn[(k0+1)*2+1] + a.z*Wn[(k0+2)*2+1] + a.w*Wn[(k0+3)*2+1];
  for (int off = 16; off; off >>= 1) { p0 += __shfl_xor(p0, off, 32); p1 += __shfl_xor(p1, off, 32); }
  p0 += bn[0]; p1 += bn[1];
  float m = fmaxf(p0, p1);
  float e0 = __expf(p0 - m), e1 = __expf(p1 - m);
  float inv = 1.f / (e0 + e1);
  float a0 = e0 * inv, a1 = e1 * inv;
  ((float4*)(xc + (size_t)i*DD))[lane] = make_float4(a.x*a0, a.y*a0, a.z*a0, a.w*a0);
  ((float4*)(xo + (size_t)i*DD))[lane] = make_float4(a.x*a1, a.y*a1, a.z*a1, a.w*a1);
}

// ---- edge attention: softmax([h[s],h[d]] @ We + be) ----
__global__ void edge_att_kernel(const float* __restrict__ h,
                                const int* __restrict__ src, const int* __restrict__ dst,
                                const float* __restrict__ We, const float* __restrict__ be,
                                float* __restrict__ ea0, float* __restrict__ ea1) {
  int e = blockIdx.x * 8 + (threadIdx.x >> 5);
  int lane = threadIdx.x & 31;
  if (e >= EE) return;
  int s = src[e], t = dst[e];
  float4 a = ((const float4*)(h + (size_t)s * DD))[lane];
  float4 b = ((const float4*)(h + (size_t)t * DD))[lane];
  int k0 = lane * 4;
  float p0 = a.x*We[(k0+0)*2+0] + a.y*We[(k0+1)*2+0] + a.z*We[(k0+2)*2+0] + a.w*We[(k0+3)*2+0]
           + b.x*We[(128+k0+0)*2+0] + b.y*We[(128+k0+1)*2+0] + b.z*We[(128+k0+2)*2+0] + b.w*We[(128+k0+3)*2+0];
  float p1 = a.x*We[(k0+0)*2+1] + a.y*We[(k0+1)*2+1] + a.z*We[(k0+2)*2+1] + a.w*We[(k0+3)*2+1]
           + b.x*We[(128+k0+0)*2+1] + b.y*We[(128+k0+1)*2+1] + b.z*We[(128+k0+2)*2+1] + b.w*We[(128+k0+3)*2+1];
  for (int off = 16; off; off >>= 1) { p0 += __shfl_xor(p0, off, 32); p1 += __shfl_xor(p1, off, 32); }
  if (lane == 0) {
    p0 += be[0]; p1 += be[1];
    float m = fmaxf(p0, p1);
    float e0 = __expf(p0 - m), e1 = __expf(p1 - m);
    float inv = 1.f / (e0 + e1);
    ea0[e] = e0 * inv; ea1[e] = e1 * inv;
  }
}

__global__ void deg_kernel(const int* __restrict__ src, const int* __restrict__ dst,
                           float* __restrict__ degout, float* __restrict__ degin) {
  int e = blockIdx.x * blockDim.x + threadIdx.x;
  if (e >= EE) return;
  atomAddF(degout + src[e], 1.f);
  atomAddF(degin  + dst[e], 1.f);
}

__global__ void dinv_kernel(float* __restrict__ dout, float* __restrict__ din) {
  int i = blockIdx.x * blockDim.x + threadIdx.x;
  if (i >= NN) return;
  dout[i] = rsqrtf(fmaxf(dout[i], 1.f));
  din[i]  = rsqrtf(fmaxf(din[i],  1.f));
}

// ---- fused 2-branch message scatter: agg[dst] += x[src]*rsqrt(deg_out[src])*ea ----
__global__ void scatter_kernel(const float* __restrict__ xc, const float* __restrict__ xo,
                               const int* __restrict__ src, const int* __restrict__ dst,
                               const float* __restrict__ ea0, const float* __restrict__ ea1,
                               const float* __restrict__ dinv_out,
                               float* __restrict__ aggc, float* __restrict__ aggo) {
  int e = blockIdx.x * 8 + (threadIdx.x >> 5);
  int lane = threadIdx.x & 31;
  if (e >= EE) return;
  int s = src[e], t = dst[e];
  float dn = dinv_out[s];
  float a0 = ea0[e] * dn, a1 = ea1[e] * dn;
  size_t sb = (size_t)s * DD, tb = (size_t)t * DD;
  #pragma unroll
  for (int j = 0; j < 4; ++j) {
    int k = lane + 32 * j;
    atomAddF(aggc + tb + k, xc[sb + k] * a0);
    atomAddF(aggo + tb + k, xo[sb + k] * a1);
  }
}

// ---- y = (agg * rsqrt(deg_in)) @ W + b, both branches, via V_WMMA_F32_16X16X4_F32 ----
__global__ void __launch_bounds__(512) gcn_gemm_kernel(
    const float* __restrict__ aggc, const float* __restrict__ aggo,
    const float* __restrict__ dinv_in,
    const float* __restrict__ Wc, const float* __restrict__ bc,
    const float* __restrict__ Wo, const float* __restrict__ bo,
    float* __restrict__ yc, float* __restrict__ yo) {
  __shared__ float As[2][16][DD];   // 16 KB: A tiles for both branches
  int r = blockIdx.x;               // row tile: rows [r*16, r*16+16)
  int tid = threadIdx.x;
  #pragma unroll
  for (int j = 0; j < 8; ++j) {
    int e = tid + j * 512;
    int mat = e >> 11, rem = e & 2047;
    int row = rem >> 7, k = rem & 127;
    int gi = r * 16 + row;
    const float* agg = mat ? aggo : aggc;
    As[mat][row][k] = agg[(size_t)gi * DD + k] * dinv_in[gi];
  }
  __syncthreads();
  int wave = tid >> 5, lane = tid & 31;
  int mat = wave >> 3, ct = wave & 7;      // 8 waves per branch, one 16x16 N-tile each
  int n0 = ct * 16, lm = lane & 15, hi = lane >> 4;
  const float* W    = mat ? Wo : Wc;
  const float* bias = mat ? bo : bc;
  float* Y          = mat ? yo : yc;
  v8f acc = {};
  for (int k = 0; k < DD; k += 4) {
    v2f av, bv;
    // A 16x4 f32 frag: lanes 0-15 hold K=k,k+1; lanes 16-31 hold K=k+2,k+3 (ISA 7.12.2)
    av.x = As[mat][lm][k + 2 * hi];
    av.y = As[mat][lm][k + 2 * hi + 1];
    // B 4x16 frag (mirrored layout assumption): VGPR0 = row K=k / K=k+2, VGPR1 = K=k+1 / K=k+3
    bv.x = W[(size_t)(k + 2 * hi)     * DD + n0 + lm];
    bv.y = W[(size_t)(k + 2 * hi + 1) * DD + n0 + lm];
    acc = __builtin_amdgcn_wmma_f32_16x16x4_f32(false, av, false, bv, (short)0, acc, false, false);
  }
  float bb = bias[n0 + lm];
  #pragma unroll
  for (int j = 0; j < 8; ++j) {
    int m = j + 8 * hi;                                   // C/D: VGPR j -> M=j (lanes<16), M=j+8 (lanes>=16)
    Y[(size_t)(r * 16 + m) * DD + n0 + lm] = acc[j] + bb;
  }
}

// ---- per-column sum / sumsq for BatchNorm (training stats) ----
__global__ void colstats_kernel(const float* __restrict__ y, float* __restrict__ s,
                                float* __restrict__ s2, int rows, int rpb) {
  int d = threadIdx.x;                  // 128 threads = columns
  int r0 = blockIdx.x * rpb;
  int r1 = min(rows, r0 + rpb);
  float a = 0.f, b = 0.f;
  for (int r = r0; r < r1; ++r) {
    float v = y[(size_t)r * DD + d];
    a += v; b += v * v;
  }
  atomAddF(s + d, a); atomAddF(s2 + d, b);
}

__global__ void bnfinal_kernel(float* __restrict__ stats,
                               const float* __restrict__ gc, const float* __restrict__ btc,
                               const float* __restrict__ go, const float* __restrict__ bto) {
  int t = threadIdx.x;                  // 256 threads: 0-127 branch c, 128-255 branch o
  int d = t & 127;
  int o = (t >> 7) ? 256 : 0;           // sum offsets
  int so = (t >> 7) ? 768 : 512;        // scale/shift offsets
  const float* g  = (t >> 7) ? go  : gc;
  const float* bt = (t >> 7) ? bto : btc;
  float mu  = stats[o + d] / (float)NN;
  float var = stats[o + 128 + d] / (float)NN - mu * mu;
  float rs  = rsqrtf(var + BNEPS);
  float sc  = rs * g[d];
  stats[so + d]       = sc;             // scale
  stats[so + 128 + d] = bt[d] - mu * sc; // shift
}

__global__ void count_kernel(const int* __restrict__ n2g, float* __restrict__ cnt) {
  int i = blockIdx.x * blockDim.x + threadIdx.x;
  if (i >= NN) return;
  atomAddF(cnt + n2g[i], 1.f);
}

// ---- normalize + ReLU + per-graph sum ----
__global__ void normscatter_kernel(const float* __restrict__ yc, const float* __restrict__ yo,
                                   const float* __restrict__ stats, const int* __restrict__ n2g,
                                   float* __restrict__ hgc, float* __restrict__ hgo) {
  size_t idx = (size_t)blockIdx.x * blockDim.x + threadIdx.x;
  if (idx >= (size_t)NN * DD) return;
  int i = (int)(idx >> 7), d = (int)(idx & 127);
  int g = n2g[i];
  float vc = fmaxf(yc[idx] * stats[512 + d] + stats[640 + d], 0.f);
  float vo = fmaxf(yo[idx] * stats[768 + d] + stats[896 + d], 0.f);
  atomAddF(hgc + (size_t)g * DD + d, vc);
  atomAddF(hgo + (size_t)g * DD + d, vo);
}

__global__ void graphdiv_kernel(float* __restrict__ hgc, float* __restrict__ hgo,
                                const float* __restrict__ cnt) {
  int idx = blockIdx.x * blockDim.x + threadIdx.x;
  if (idx >= GG * DD) return;
  float c = 1.f / fmaxf(cnt[idx >> 7], 1.f);
  hgc[idx] *= c; hgo[idx] *= c;
}

__global__ void meanc_kernel(const float* __restrict__ hgc, float* __restrict__ meanc) {
  int d = threadIdx.x;
  float s = 0.f;
  for (int g = 0; g < GG; ++g) s += hgc[(size_t)g * DD + d];
  meanc[d] = s * (1.f / (float)GG);
}

__global__ void hgco_kernel(const float* __restrict__ meanc, const float* __restrict__ hgo,
                            float* __restrict__ hgco) {
  int idx = blockIdx.x * blockDim.x + threadIdx.x;
  if (idx >= GG * DD) return;
  hgco[idx] = meanc[idx & 127] + hgo[idx];
}

struct ReadoutArgs {
  const float* x[3];
  const float *W0[3], *b0[3], *W1[3], *b1[3], *W2[3], *b2[3];
  float* out;
};

// ---- MLPReadout: 128 -> 64 -> 32 -> 10, one block per (graph, head) ----
__global__ void mlp_readout_kernel(ReadoutArgs a) {
  int g = blockIdx.x, head = blockIdx.y, t = threadIdx.x;   // 64 threads
  __shared__ float xb[128], z1[64], z2[32];
  const float* x = a.x[head];
  xb[t] = x[(size_t)g * DD + t];
  xb[t + 64] = x[(size_t)g * DD + t + 64];
  __syncthreads();
  const float* W0 = a.W0[head]; const float* b0 = a.b0[head];
  float s = b0[t];
  for (int k = 0; k < 128; ++k) s += xb[k] * W0[k * 64 + t];
  z1[t] = fmaxf(s, 0.f);
  __syncthreads();
  if (t < 32) {
    const float* W1 = a.W1[head]; const float* b1 = a.b1[head];
    float u = b1[t];
    for (int k = 0; k < 64; ++k) u += z1[k] * W1[k * 32 + t];
    z2[t] = fmaxf(u, 0.f);
  }
  __syncthreads();
  if (t < OUTD) {
    const float* W2 = a.W2[head]; const float* b2 = a.b2[head];
    float u = b2[t];
    for (int k = 0; k < 32; ++k) u += z2[k] * W2[k * OUTD + t];
    a.out[head * GG * OUTD + g * OUTD + t] = u;
  }
}

extern "C" void kernel_launch(void* const* d_in, const int* in_sizes, int n_in,
                              void* d_out, int out_size, void* d_ws, size_t ws_size,
                              hipStream_t stream) {
  // Flattened input order (setup_inputs insertion order, params leaves depth-first):
  const float* h    = (const float*)d_in[0];
  const int*   src  = (const int*)d_in[1];
  const int*   dst  = (const int*)d_in[2];
  const int*   n2g  = (const int*)d_in[3];
  const float* Wn   = (const float*)d_in[4];
  const float* bn   = (const float*)d_in[5];
  const float* We   = (const float*)d_in[6];
  const float* be   = (const float*)d_in[7];
  const float* Wc   = (const float*)d_in[8];
  const float* bc   = (const float*)d_in[9];
  const float* gmc  = (const float*)d_in[10];
  const float* btc  = (const float*)d_in[11];
  const float* Wo   = (const float*)d_in[12];
  const float* bo   = (const float*)d_in[13];
  const float* gmo  = (const float*)d_in[14];
  const float* bto  = (const float*)d_in[15];

  float* w = (float*)d_ws;
  const size_t ND = (size_t)NN * DD;
  float* xc    = w;                 // reused as yc after scatter
  float* xo    = w + ND;            // reused as yo after scatter
  float* aggc  = w + 2 * ND;
  float* aggo  = w + 3 * ND;
  float* ea0   = w + 4 * ND;
  float* ea1   = ea0 + EE;
  float* dout  = ea1 + EE;          // deg_out counts -> rsqrt in place
  float* din   = dout + NN;         // deg_in  counts -> rsqrt in place
  float* stats = din + NN;          // 1024: sumc,sqc,sumo,sqo,scalec,shiftc,scaleo,shifto
  float* hgc   = stats + 1024;
  float* hgo   = hgc + (size_t)GG * DD;
  float* hgco  = hgo + (size_t)GG * DD;
  float* meanc = hgco + (size_t)GG * DD;
  float* cnt   = meanc + DD;
  float* yc = xc; float* yo = xo;

  // zero accumulators: [aggc..aggo) and the tail region [dout..cnt+GG)
  long long z1n = 2LL * (long long)ND;
  zerof_kernel<<<(int)((z1n + 255) / 256), 256, 0, stream>>>(aggc, z1n);
  long long z2n = 2LL * NN + 1024 + 3LL * GG * DD + DD + GG;
  zerof_kernel<<<(int)((z2n + 255) / 256), 256, 0, stream>>>(dout, z2n);

  node_att_kernel<<<NN / 8, 256, 0, stream>>>(h, Wn, bn, xc, xo);
  edge_att_kernel<<<EE / 8, 256, 0, stream>>>(h, src, dst, We, be, ea0, ea1);
  deg_kernel<<<(EE + 255) / 256, 256, 0, stream>>>(src, dst, dout, din);
  dinv_kernel<<<(NN + 255) / 256, 256, 0, stream>>>(dout, din);
  scatter_kernel<<<EE / 8, 256, 0, stream>>>(xc, xo, src, dst, ea0, ea1, dout, aggc, aggo);

  gcn_gemm_kernel<<<NN / 16, 512, 0, stream>>>(aggc, aggo, din, Wc, bc, Wo, bo, yc, yo);

  colstats_kernel<<<(NN + 511) / 512, DD, 0, stream>>>(yc, stats,       stats + 128, NN, 512);
  colstats_kernel<<<(NN + 511) / 512, DD, 0, stream>>>(yo, stats + 256, stats + 384, NN, 512);
  bnfinal_kernel<<<1, 256, 0, stream>>>(stats, gmc, btc, gmo, bto);

  count_kernel<<<(NN + 255) / 256, 256, 0, stream>>>(n2g, cnt);
  normscatter_kernel<<<(int)((ND + 255) / 256), 256, 0, stream>>>(yc, yo, stats, n2g, hgc, hgo);
  graphdiv_kernel<<<(GG * DD) / 256, 256, 0, stream>>>(hgc, hgo, cnt);
  meanc_kernel<<<1, DD, 0, stream>>>(hgc, meanc);
  hgco_kernel<<<(GG * DD) / 256, 256, 0, stream>>>(meanc, hgo, hgco);

  ReadoutArgs ra;
  ra.x[0] = hgc; ra.x[1] = hgo; ra.x[2] = hgco;
  for (int hsel = 0; hsel < 3; ++hsel) {
    int base = 16 + hsel * 6;
    ra.W0[hsel] = (const float*)d_in[base + 0];
    ra.b0[hsel] = (const float*)d_in[base + 1];
    ra.W1[hsel] = (const float*)d_in[base + 2];
    ra.b1[hsel] = (const float*)d_in[base + 3];
    ra.W2[hsel] = (const float*)d_in[base + 4];
    ra.b2[hsel] = (const float*)d_in[base + 5];
  }
  ra.out = (float*)d_out;
  mlp_readout_kernel<<<dim3(GG, 3), 64, 0, stream>>>(ra);
}